// VectorQuantizedVAE_23063974379562
// MI455X (gfx1250) — compile-verified
//
#include <hip/hip_runtime.h>

// Problem sizes (fixed by the reference).
#define DDIM  256
#define KCODE 1024
#define NROWS 65536          // 64*32*32
#define NTILE (KCODE / 16)   // 64 code tiles
#define WPAD  260            // padded LDS row stride (floats) -> conflict-free B reads

typedef float v2f __attribute__((ext_vector_type(2)));
typedef float v4f __attribute__((ext_vector_type(4)));
typedef float v8f __attribute__((ext_vector_type(8)));
typedef int   v4i __attribute__((vector_size(16)));   // matches builtin proto

// ---------------------------------------------------------------------------
// CDNA5 async global->LDS path (ASYNCcnt), guarded so the build can't break.
// ---------------------------------------------------------------------------
#if defined(__AMDGCN__) && __has_builtin(__builtin_amdgcn_global_load_async_to_lds_b128)
#define HAVE_ASYNC 1
#else
#define HAVE_ASYNC 0
#endif

#if HAVE_ASYNC
typedef __attribute__((address_space(1))) v4i g_v4i;
typedef __attribute__((address_space(3))) v4i l_v4i;

__device__ __forceinline__ void async_wait_le4() {
#if __has_builtin(__builtin_amdgcn_s_wait_asynccnt)
    __builtin_amdgcn_s_wait_asynccnt(4);
#else
    asm volatile("s_wait_asynccnt 0x4" ::: "memory");
#endif
}
__device__ __forceinline__ void async_wait_0() {
#if __has_builtin(__builtin_amdgcn_s_wait_asynccnt)
    __builtin_amdgcn_s_wait_asynccnt(0);
#else
    asm volatile("s_wait_asynccnt 0x0" ::: "memory");
#endif
}

// Each of the 256 threads copies 4 x 16B of the 16KB tile straight into LDS.
__device__ __forceinline__ void stage_tile_async(const float* __restrict__ weight,
                                                 int ct, float* buf, int tid) {
    const char* gbase = (const char*)(weight + (size_t)ct * 16 * DDIM);
#pragma unroll
    for (int i = 0; i < 4; ++i) {
        int idx = tid + i * 256;           // float4 element id, 0..1023
        int row = idx >> 6;                // 64 float4 per code row
        int c4  = idx & 63;
        g_v4i* gp = (g_v4i*)(gbase + (size_t)idx * 16);
        l_v4i* lp = (l_v4i*)((char*)buf + (size_t)(row * WPAD + c4 * 4) * 4);
        __builtin_amdgcn_global_load_async_to_lds_b128(gp, lp, 0, 0);
    }
}
#endif

// ---------------------------------------------------------------------------
// Kernel 1: sqr[k] = sum_j weight[k][j]^2
// ---------------------------------------------------------------------------
__global__ __launch_bounds__(256)
void vq_sqr_kernel(const float* __restrict__ weight, float* __restrict__ sqr) {
    int k = blockIdx.x * blockDim.x + threadIdx.x;     // 0..1023
    const v4f* wr = (const v4f*)(weight + (size_t)k * DDIM);
    float s = 0.0f;
#pragma unroll 8
    for (int j = 0; j < DDIM / 4; ++j) {
        v4f v = wr[j];
        s += v.x * v.x + v.y * v.y + v.z * v.z + v.w * v.w;
    }
    sqr[k] = s;
}

// ---------------------------------------------------------------------------
// Kernel 2: per 16-row tile, argmin_k (sqr[k] - 2 * <z, w_k>) via f32 WMMA,
// double-buffered async weight staging, then gather winning code rows.
// Block = 256 threads (8 waves), each wave owns 16 rows -> 128 rows / block.
// ---------------------------------------------------------------------------
__global__ __launch_bounds__(256)
void vq_main_kernel(const float* __restrict__ input,
                    const float* __restrict__ weight,
                    const float* __restrict__ sqr,
                    float* __restrict__ out_values,
                    float* __restrict__ out_indices,
                    float* __restrict__ out_vectors) {
    __shared__ float wlds[2][16 * WPAD];  // double-buffered 16x256 weight tiles
    __shared__ float sqlds[KCODE];        // all 1024 squared norms

    const int tid  = threadIdx.x;
    const int lane = tid & 31;
    const int wave = tid >> 5;            // 0..7
    const int hh   = lane >> 4;           // lane half: 0 or 1
    const int l16  = lane & 15;

    const int row_base = blockIdx.x * 128 + wave * 16;

    // Stage all squared norms once (4 KB).
    for (int i = tid; i < KCODE; i += 256) sqlds[i] = sqr[i];

    // Preload the wave's 16x256 A tile as 64 f32-WMMA A fragments (NT: read once).
    // f32 16x4 A layout: lane -> M = lane&15; VGPR j holds K = 4*j + 2*half + {0,1}.
    v2f afrag[64];
    {
        const float* arow = input + (size_t)(row_base + l16) * DDIM + 2 * hh;
#pragma unroll
        for (int j = 0; j < 64; ++j) {
            afrag[j] = __builtin_nontemporal_load((const v2f*)(arow + 4 * j));
        }
    }

    float best[8];
    int   bidx[8];
#pragma unroll
    for (int r = 0; r < 8; ++r) { best[r] = 3.402823e38f; bidx[r] = 0; }

#if HAVE_ASYNC
    stage_tile_async(weight, 0, wlds[0], tid);   // prologue: tile 0 in flight
#endif

    for (int ct = 0; ct < NTILE; ++ct) {
        const int cur = ct & 1;
#if HAVE_ASYNC
        // Kick off next tile into the other buffer (freed by last iteration's
        // trailing barrier), then wait for the current tile's 4 in-order ops.
        if (ct + 1 < NTILE) {
            stage_tile_async(weight, ct + 1, wlds[cur ^ 1], tid);
            async_wait_le4();
        } else {
            async_wait_0();
        }
        __syncthreads();                   // LDS tile visible to all waves
#else
        __syncthreads();
        {   // synchronous fallback staging, coalesced float4
            const v4f* wsrc = (const v4f*)(weight + (size_t)ct * 16 * DDIM);
#pragma unroll
            for (int i = 0; i < 4; ++i) {
                int idx = tid + i * 256;
                int row = idx >> 6;
                int c4  = idx & 63;
                *(v4f*)(&wlds[cur][row * WPAD + c4 * 4]) = wsrc[row * 64 + c4];
            }
        }
        __syncthreads();
#endif

        // 16x16 cov tile over D=256 in 64 steps of K=4, two accumulators.
        v8f acc0 = {};
        v8f acc1 = {};
        const float* brow = &wlds[cur][l16 * WPAD + 2 * hh];
#pragma unroll
        for (int j = 0; j < 64; j += 2) {
            v2f b0 = *(const v2f*)(brow + 4 * j);
            v2f b1 = *(const v2f*)(brow + 4 * (j + 1));
            acc0 = __builtin_amdgcn_wmma_f32_16x16x4_f32(
                false, afrag[j],     false, b0, (short)0, acc0, false, false);
            acc1 = __builtin_amdgcn_wmma_f32_16x16x4_f32(
                false, afrag[j + 1], false, b1, (short)0, acc1, false, false);
        }

        // C layout: element r, lane -> (M = r + 8*half, N = lane&15).
        const int   code = ct * 16 + l16;
        const float sq   = sqlds[code];
#pragma unroll
        for (int r = 0; r < 8; ++r) {
            float dist = sq - 2.0f * (acc0[r] + acc1[r]);
            if (dist < best[r]) { best[r] = dist; bidx[r] = code; }
        }

#if HAVE_ASYNC
        __syncthreads();                   // reads done: buffer reusable next iter
#endif
    }

    // Min-reduce across the 16 lanes sharing each row (xor < 16 stays in half).
#pragma unroll
    for (int r = 0; r < 8; ++r) {
        for (int off = 8; off >= 1; off >>= 1) {
            float ov = __shfl_xor(best[r], off, 32);
            int   oi = __shfl_xor(bidx[r], off, 32);
            if (ov < best[r] || (ov == best[r] && oi < bidx[r])) {
                best[r] = ov; bidx[r] = oi;
            }
        }
    }

    // Lanes 0 and 16 now hold winners for rows r and r+8 respectively.
    if (l16 == 0) {
#pragma unroll
        for (int r = 0; r < 8; ++r) {
            out_indices[row_base + hh * 8 + r] = (float)bidx[r];
        }
    }

    // Gather winning code rows (codebook stays hot in L2); NT streaming stores.
#pragma unroll
    for (int m = 0; m < 16; ++m) {
        int idx = __shfl(bidx[m & 7], (m >> 3) * 16, 32);
        const v4f* src = (const v4f*)(weight + (size_t)idx * DDIM);
        size_t obase = (size_t)(row_base + m) * DDIM;
#pragma unroll
        for (int j = 0; j < 2; ++j) {
            int c4 = lane + j * 32;            // float4 column 0..63
            v4f v = src[c4];
            __builtin_nontemporal_store(v, (v4f*)(out_values  + obase + c4 * 4));
            __builtin_nontemporal_store(v, (v4f*)(out_vectors + obase + c4 * 4));
        }
    }
}

// ---------------------------------------------------------------------------
// Launch wrapper
// ---------------------------------------------------------------------------
extern "C" void kernel_launch(void* const* d_in, const int* in_sizes, int n_in,
                              void* d_out, int out_size, void* d_ws, size_t ws_size,
                              hipStream_t stream) {
    const float* input  = (const float*)d_in[0];   // [65536, 256]
    const float* weight = (const float*)d_in[1];   // [1024, 256]

    float* sqr = (float*)d_ws;                     // [1024] scratch

    float* out_values  = (float*)d_out;                         // [65536*256]
    float* out_indices = out_values + (size_t)NROWS * DDIM;     // [65536]
    float* out_vectors = out_indices + NROWS;                   // [65536*256]

    vq_sqr_kernel<<<KCODE / 256, 256, 0, stream>>>(weight, sqr);
    vq_main_kernel<<<NROWS / 128, 256, 0, stream>>>(
        input, weight, sqr, out_values, out_indices, out_vectors);
}